// DynamicGNN_74036646248564
// MI455X (gfx1250) — compile-verified
//
#include <hip/hip_runtime.h>
#include <hip/hip_bf16.h>

#define N_NODES   50000
#define N_EDGES   1250000
#define IN_DIM    5
#define OUT_DIM   64
#define NUM_REL   3
#define NUM_LAYERS 3
#define LN_EPS    1e-5f

typedef __attribute__((ext_vector_type(2))) float v2f;
typedef __attribute__((ext_vector_type(8))) float v8f;

// ---------------------------------------------------------------------------
// counts / inv_denom
// ---------------------------------------------------------------------------
__global__ void zero_i32_kernel(int* __restrict__ p, int n) {
    int i = blockIdx.x * blockDim.x + threadIdx.x;
    if (i < n) p[i] = 0;
}

__global__ void count_kernel(const int* __restrict__ dst,
                             const int* __restrict__ etype,
                             int* __restrict__ counts, int nE) {
    int e = blockIdx.x * blockDim.x + threadIdx.x;
    if (e < nE) atomicAdd(&counts[dst[e] * NUM_REL + etype[e]], 1);
}

__global__ void inv_denom_kernel(const int* __restrict__ counts,
                                 float* __restrict__ inv, int n) {
    int i = blockIdx.x * blockDim.x + threadIdx.x;
    if (i < n) {
        int c = counts[i];
        inv[i] = 1.0f / (float)(c > 1 ? c : 1);
    }
}

// ---------------------------------------------------------------------------
// input projection: h = x @ feat2c_W + b   ([N,5]·[5,64])
// ---------------------------------------------------------------------------
__global__ void feat2c_kernel(const float* __restrict__ x,
                              const float* __restrict__ W,
                              const float* __restrict__ b,
                              float* __restrict__ h, int n) {
    int idx = blockIdx.x * blockDim.x + threadIdx.x;
    if (idx >= n * OUT_DIM) return;
    int node = idx >> 6;
    int o    = idx & 63;
    float acc = b[o];
#pragma unroll
    for (int d = 0; d < IN_DIM; ++d)
        acc += x[node * IN_DIM + d] * W[d * OUT_DIM + o];
    h[idx] = acc;
}

// ---------------------------------------------------------------------------
// WMMA transform: hr[n,r,:] = h[n,:] @ relW[r]  (r=0..2, waves 0..2)
//                 out[n,:]  = h[n,:] @ rootW + bias  (wave 3)
// One block = 16 node rows. 4 waves, each owning one 64x64 weight matrix.
// Uses V_WMMA_F32_16X16X4_F32 (exact fp32 matrix math).
// ---------------------------------------------------------------------------
__global__ __launch_bounds__(128)
void rgcn_transform_kernel(const float* __restrict__ h,
                           const float* __restrict__ relW,   // [3][64][64]
                           const float* __restrict__ rootW,  // [64][64]
                           const float* __restrict__ bias,   // [64]
                           float* __restrict__ hr,           // [N][3][64]
                           float* __restrict__ outbuf)       // [N][64]
{
    __shared__ float lh[16][65];   // padded: bank-conflict-free column reads
    const int row0 = blockIdx.x * 16;

    // stage 16x64 tile of h into LDS
    for (int i = threadIdx.x; i < 16 * 64; i += 128) {
        int r = i >> 6, c = i & 63;
        lh[r][c] = h[(size_t)(row0 + r) * OUT_DIM + c];
    }
    __syncthreads();

    const int wave = threadIdx.x >> 5;          // 0..3 -> weight matrix
    const int lane = threadIdx.x & 31;
    const int mrow = lane & 15;                 // A: M row / B: N col (in tile)
    const int khalf = (lane >> 4) << 1;         // lanes 0-15: K{0,1}; 16-31: K{2,3}

    const float* __restrict__ W =
        (wave < NUM_REL) ? (relW + (size_t)wave * OUT_DIM * OUT_DIM) : rootW;

    v8f acc[4];
#pragma unroll
    for (int nt = 0; nt < 4; ++nt) acc[nt] = (v8f){0.f,0.f,0.f,0.f,0.f,0.f,0.f,0.f};

#pragma unroll
    for (int k0 = 0; k0 < OUT_DIM; k0 += 4) {
        const int kk = k0 + khalf;
        v2f a;
        a.x = lh[mrow][kk];
        a.y = lh[mrow][kk + 1];
        const float* __restrict__ Wk0 = W + (size_t)kk * OUT_DIM;
        const float* __restrict__ Wk1 = Wk0 + OUT_DIM;
#pragma unroll
        for (int nt = 0; nt < 4; ++nt) {
            v2f b;
            b.x = Wk0[nt * 16 + mrow];
            b.y = Wk1[nt * 16 + mrow];
            // (neg_a, A, neg_b, B, c_mod, C, reuse_a, reuse_b)
            acc[nt] = __builtin_amdgcn_wmma_f32_16x16x4_f32(
                false, a, false, b, (short)0, acc[nt], false, false);
        }
    }

    // D layout: lane%16 = N col, VGPR v -> M = v + (lane>=16 ? 8 : 0)
    const int rbase = row0 + ((lane >> 4) << 3);
    if (wave < NUM_REL) {
#pragma unroll
        for (int nt = 0; nt < 4; ++nt) {
            const int col = nt * 16 + mrow;
#pragma unroll
            for (int v = 0; v < 8; ++v) {
                int row = rbase + v;
                hr[((size_t)row * NUM_REL + wave) * OUT_DIM + col] = acc[nt][v];
            }
        }
    } else {
#pragma unroll
        for (int nt = 0; nt < 4; ++nt) {
            const int col = nt * 16 + mrow;
            const float bs = bias[col];
#pragma unroll
            for (int v = 0; v < 8; ++v) {
                int row = rbase + v;
                outbuf[(size_t)row * OUT_DIM + col] = acc[nt][v] + bs;
            }
        }
    }
}

// ---------------------------------------------------------------------------
// edge scatter: out[dst] += hr[src, rel] * inv_denom[dst, rel]
// One wave32 per edge; float2 per lane -> coalesced 256B row read.
// ---------------------------------------------------------------------------
__global__ __launch_bounds__(256)
void edge_scatter_kernel(const int* __restrict__ src,
                         const int* __restrict__ dst,
                         const int* __restrict__ etype,
                         const float* __restrict__ hr,
                         const float* __restrict__ inv_denom,
                         float* __restrict__ outbuf, int nE) {
    int e = blockIdx.x * 8 + (threadIdx.x >> 5);
    if (e >= nE) return;
    const int lane = threadIdx.x & 31;
    const int s = src[e];
    const int d = dst[e];
    const int r = etype[e];
    const float inv = inv_denom[d * NUM_REL + r];
    const float2 m =
        ((const float2*)(hr + ((size_t)s * NUM_REL + r) * OUT_DIM))[lane];
    float* o = outbuf + (size_t)d * OUT_DIM + lane * 2;
    unsafeAtomicAdd(o,     m.x * inv);
    unsafeAtomicAdd(o + 1, m.y * inv);
}

// ---------------------------------------------------------------------------
// finalize: h = LayerNorm(ReLU(out)) ; one wave32 per node, shfl reductions
// ---------------------------------------------------------------------------
__global__ __launch_bounds__(256)
void finalize_kernel(const float* __restrict__ outbuf,
                     const float* __restrict__ gamma,
                     const float* __restrict__ beta,
                     float* __restrict__ hout, int n) {
    int node = blockIdx.x * 8 + (threadIdx.x >> 5);
    if (node >= n) return;
    const int lane = threadIdx.x & 31;
    float2 v = ((const float2*)(outbuf + (size_t)node * OUT_DIM))[lane];
    v.x = fmaxf(v.x, 0.0f);
    v.y = fmaxf(v.y, 0.0f);
    float s  = v.x + v.y;
    float sq = v.x * v.x + v.y * v.y;
#pragma unroll
    for (int off = 16; off > 0; off >>= 1) {
        s  += __shfl_xor(s,  off, 32);
        sq += __shfl_xor(sq, off, 32);
    }
    const float mu   = s * (1.0f / OUT_DIM);
    const float var  = sq * (1.0f / OUT_DIM) - mu * mu;
    const float rstd = rsqrtf(var + LN_EPS);
    const int c0 = lane * 2;
    float2 o;
    o.x = (v.x - mu) * rstd * gamma[c0]     + beta[c0];
    o.y = (v.y - mu) * rstd * gamma[c0 + 1] + beta[c0 + 1];
    ((float2*)(hout + (size_t)node * OUT_DIM))[lane] = o;
}

// ---------------------------------------------------------------------------
extern "C" void kernel_launch(void* const* d_in, const int* in_sizes, int n_in,
                              void* d_out, int out_size, void* d_ws, size_t ws_size,
                              hipStream_t stream) {
    const float* x      = (const float*)d_in[0];
    const int*   ei     = (const int*)  d_in[1];   // [2,E]
    const int*   src    = ei;
    const int*   dst    = ei + N_EDGES;
    const int*   etype  = (const int*)  d_in[2];
    const float* f2cW   = (const float*)d_in[3];
    const float* f2cb   = (const float*)d_in[4];
    const float* rgW    = (const float*)d_in[5];   // [3][3][64][64]
    const float* rgRoot = (const float*)d_in[6];   // [3][64][64]
    const float* rgBias = (const float*)d_in[7];   // [3][64]
    const float* gamma  = (const float*)d_in[8];
    const float* beta   = (const float*)d_in[9];

    float* ws = (float*)d_ws;
    float* h_a       = ws;  ws += (size_t)N_NODES * OUT_DIM;
    float* h_b       = ws;  ws += (size_t)N_NODES * OUT_DIM;
    float* hr        = ws;  ws += (size_t)N_NODES * NUM_REL * OUT_DIM;
    float* outbuf    = ws;  ws += (size_t)N_NODES * OUT_DIM;
    float* inv_denom = ws;  ws += (size_t)N_NODES * NUM_REL;
    int*   counts    = (int*)ws;

    const int NR = N_NODES * NUM_REL;

    zero_i32_kernel<<<(NR + 255) / 256, 256, 0, stream>>>(counts, NR);
    count_kernel<<<(N_EDGES + 255) / 256, 256, 0, stream>>>(dst, etype, counts, N_EDGES);
    inv_denom_kernel<<<(NR + 255) / 256, 256, 0, stream>>>(counts, inv_denom, NR);
    feat2c_kernel<<<((size_t)N_NODES * OUT_DIM + 255) / 256, 256, 0, stream>>>(
        x, f2cW, f2cb, h_a, N_NODES);

    float* hcur = h_a;
    float* hnext = h_b;
    for (int l = 0; l < NUM_LAYERS; ++l) {
        rgcn_transform_kernel<<<N_NODES / 16, 128, 0, stream>>>(
            hcur,
            rgW    + (size_t)l * NUM_REL * OUT_DIM * OUT_DIM,
            rgRoot + (size_t)l * OUT_DIM * OUT_DIM,
            rgBias + (size_t)l * OUT_DIM,
            hr, outbuf);
        edge_scatter_kernel<<<(N_EDGES + 7) / 8, 256, 0, stream>>>(
            src, dst, etype, hr, inv_denom, outbuf, N_EDGES);
        float* ho = (l == NUM_LAYERS - 1) ? (float*)d_out : hnext;
        finalize_kernel<<<(N_NODES + 7) / 8, 256, 0, stream>>>(
            outbuf, gamma, beta, ho, N_NODES);
        float* t = hcur; hcur = hnext; hnext = t;
    }
}